// FASTFeatureExtractor_82952998355298
// MI455X (gfx1250) — compile-verified
//
#include <hip/hip_runtime.h>
#include <hip/hip_bf16.h>
#include <stdint.h>

#define IMG_H 1024
#define IMG_W 1024
#define NB    4
#define TILE  32
#define HALO  4                      // 3 (FAST circle) + 1 (NMS ring)
#define IN_T  (TILE + 2 * HALO)      // 40  (40 floats = 10 x b128 per row)
#define MD_T  (TILE + 2)             // 34
#define THRESH (11.0f / 255.0f)

// Low 32 bits of a flat shared-aperture address are the LDS byte offset
// (aperture lives in addr[63:32] on gfx1250).
__device__ __forceinline__ uint32_t lds_off_u32(const void* p) {
  return (uint32_t)(uintptr_t)p;
}

// CDNA5 async DMA: global -> LDS, 16 bytes/lane, GVS mode
// (SGPR64 base + VGPR32 byte offset). Tracked by ASYNCcnt.
__device__ __forceinline__ void async_ld_b128(uint32_t ldsoff, const float* base,
                                              int byteoff) {
  asm volatile("global_load_async_to_lds_b128 %0, %1, %2"
               :: "v"(ldsoff), "v"(byteoff), "s"(base)
               : "memory");
}

__device__ __forceinline__ void wait_async0() {
  asm volatile("s_wait_asynccnt 0" ::: "memory");
}

// True iff the 16-bit pattern has a circular run of >= 9 ones.
// Replaces the 65536-entry LUT (two random gathers/pixel) with 5 VALU ops.
__device__ __forceinline__ bool run_ge_9(uint32_t m) {
  uint32_t x = m | (m << 16);   // circular doubling
  x &= x >> 1;                  // runs >= 2
  x &= x >> 2;                  // runs >= 4
  x &= x >> 4;                  // runs >= 8
  x &= x >> 1;                  // runs >= 9
  return (x & 0xFFFFu) != 0u;
}

__global__ __launch_bounds__(256)
void fast_fused_kernel(const float* __restrict__ x,
                       float* __restrict__ out_md,
                       float* __restrict__ out_feat) {
  __shared__ float s_in[IN_T][IN_T];   // 6400 B input window
  __shared__ float s_md[MD_T][MD_T];   // 4624 B mean-diff tile (+1 NMS ring)

  const int tx = threadIdx.x;          // 0..31
  const int ty = threadIdx.y;          // 0..7
  const int tid = ty * 32 + tx;
  const int x0 = blockIdx.x * TILE;
  const int y0 = blockIdx.y * TILE;
  const int b  = blockIdx.z;
  const float* base = x + (size_t)b * (IMG_H * IMG_W);

  // 40x40 load window, shifted (not clamped per-element) to stay in-image.
  // ox stays a multiple of 4 -> every row is 10 aligned b128 chunks.
  const int oy = min(max(y0 - HALO, 0), IMG_H - IN_T);
  const int ox = min(max(x0 - HALO, 0), IMG_W - IN_T);

  // FAST-16 circle offsets, radius 3 (same circular order as reference).
  const int di[16] = {-3,-3,-2,-1, 0, 1, 2, 3, 3, 3, 2, 1, 0,-1,-2,-3};
  const int dj[16] = { 0, 1, 2, 3, 3, 3, 2, 1, 0,-1,-2,-3,-3,-3,-2,-1};

  // ---- Phase 1: async-DMA the 40x40 window into LDS (400 x b128) ----
  for (int idx = tid; idx < IN_T * (IN_T / 4); idx += 256) {
    const int r  = idx / (IN_T / 4);
    const int c4 = (idx % (IN_T / 4)) * 4;
    async_ld_b128(lds_off_u32(&s_in[r][c4]), base,
                  ((oy + r) * IMG_W + ox + c4) * 4);
  }
  wait_async0();       // this wave's async LDS writes complete
  __syncthreads();     // cross-wave visibility

  // ---- Phase 2: 34x34 mean-|diff| tile (zero outside valid region) ----
  const int ry0 = y0 - 1 - oy;         // s_in row of md[0][*]  (interior: 3)
  const int rx0 = x0 - 1 - ox;         // s_in col of md[*][0]  (interior: 3)
  for (int idx = tid; idx < MD_T * MD_T; idx += 256) {
    const int r = idx / MD_T, c = idx % MD_T;
    const int gy = y0 - 1 + r, gx = x0 - 1 + c;
    float md = 0.0f;
    if (gy >= 3 && gy <= IMG_H - 4 && gx >= 3 && gx <= IMG_W - 4) {
      const float ctr = s_in[ry0 + r][rx0 + c];
      float s = 0.0f;
#pragma unroll
      for (int k = 0; k < 16; ++k)
        s += fabsf(s_in[ry0 + r + di[k]][rx0 + c + dj[k]] - ctr);
      md = s * (1.0f / 16.0f);
    }
    s_md[r][c] = md;
  }
  __syncthreads();

  // ---- Phase 3: NMS + corner test + stores (4 pixels / thread) ----
  const int cy0 = y0 - oy;             // s_in row of output row py=0 (interior: 4)
  const int cx0 = x0 - ox;
#pragma unroll
  for (int i = 0; i < 4; ++i) {
    const int py = ty + i * 8;          // 0..31
    const int px = tx;                  // 0..31
    const int gy = y0 + py, gx = x0 + px;
    const int mr = py + 1, mc = px + 1;

    const float md_c = s_md[mr][mc];
    float nb = s_md[mr - 1][mc - 1];
    nb = fmaxf(nb, s_md[mr - 1][mc    ]);
    nb = fmaxf(nb, s_md[mr - 1][mc + 1]);
    nb = fmaxf(nb, s_md[mr    ][mc - 1]);
    nb = fmaxf(nb, s_md[mr    ][mc + 1]);
    nb = fmaxf(nb, s_md[mr + 1][mc - 1]);
    nb = fmaxf(nb, s_md[mr + 1][mc    ]);
    nb = fmaxf(nb, s_md[mr + 1][mc + 1]);

    float feat = 0.0f;
    if (gy >= 3 && gy <= IMG_H - 4 && gx >= 3 && gx <= IMG_W - 4 &&
        md_c >= nb) {                   // pooled == md  <=>  md >= max(neighbors)
      const float ctr = s_in[cy0 + py][cx0 + px];
      uint32_t pos = 0u, neg = 0u;
#pragma unroll
      for (int k = 0; k < 16; ++k) {
        const float v = s_in[cy0 + py + di[k]][cx0 + px + dj[k]];
        pos |= (v >= ctr + THRESH) ? (1u << (15 - k)) : 0u;
        neg |= (v <= ctr - THRESH) ? (1u << (15 - k)) : 0u;
      }
      if (run_ge_9(pos) || run_ge_9(neg)) feat = 1.0f;
    }

    const size_t o = (size_t)b * (IMG_H * IMG_W) + (size_t)gy * IMG_W + gx;
    out_md[o]   = md_c;
    out_feat[o] = feat;
  }
}

extern "C" void kernel_launch(void* const* d_in, const int* in_sizes, int n_in,
                              void* d_out, int out_size, void* d_ws, size_t ws_size,
                              hipStream_t stream) {
  (void)in_sizes; (void)n_in; (void)d_ws; (void)ws_size; (void)out_size;
  const float* x = (const float*)d_in[0];
  // d_in[1] (65536-entry LUT) is intentionally unused: replaced by bit ops.
  float* out_md   = (float*)d_out;
  float* out_feat = (float*)d_out + (size_t)NB * IMG_H * IMG_W;

  dim3 grid(IMG_W / TILE, IMG_H / TILE, NB);
  dim3 block(32, 8, 1);
  hipLaunchKernelGGL(fast_fused_kernel, grid, block, 0, stream,
                     x, out_md, out_feat);
}